// SinkhornLoss_35364760715833
// MI455X (gfx1250) — compile-verified
//
#include <hip/hip_runtime.h>
#include <math.h>

// ---------------------------------------------------------------------------
// Sinkhorn loss (geomloss-style, p=2, blur=0.05, scaling=0.5, K=16) for
// B=16 batches of N=1536 1-D points, uniform weights.
//
// Roofline: ~1.36e9 exp evaluations vs ~400 KB HBM traffic => purely
// transcendental/VALU bound. Strategy:
//  - column vectors (points + scaled potentials) live in LDS, staged with
//    CDNA5 async global->LDS DMA (ASYNCcnt path) + s_wait_asynccnt
//  - softmin evaluated in log2 domain so the hot loop is a raw v_exp_f32
//  - 4 independent max/sum accumulators break serial FP chains
//  - one kernel launch per Sinkhorn iteration (graph-captured) = global sync
//  - final reduction uses V_WMMA_F32_16X16X4_F32 x ones as a 64-way adder
// ---------------------------------------------------------------------------

#define BATCH   16
#define NPTS    1536
#define BN      (BATCH * NPTS)
#define BLOCK   256
#define RB      6                   // 6 row-blocks * 256 rows = 1536 rows
#define NLOG2   (-10.58496250f)     // -log2(1536)
#define LN2     0.69314718056f
#define LOG2E   1.44269504089f
#define BLUR2   0.0025f             // blur^2 = eps floor
#define EPS_STRIDE 32               // per-batch eps schedule stride (18 used)

typedef float v2f __attribute__((ext_vector_type(2)));
typedef float v8f __attribute__((ext_vector_type(8)));

// ---------------------------------------------------------------------------
// Kernel 1: per-batch diameter -> eps annealing schedule; zero f/g potentials.
// ---------------------------------------------------------------------------
__global__ __launch_bounds__(BLOCK)
void sink_setup(const float* __restrict__ x, const float* __restrict__ y,
                float* __restrict__ f0, float* __restrict__ g0,
                float* __restrict__ eps_sched)
{
    const int b   = blockIdx.x;
    const int tid = threadIdx.x;
    __shared__ float smn[BLOCK];
    __shared__ float smx[BLOCK];

    float mn =  3.0e38f, mx = -3.0e38f;
    const float* xb = x + b * NPTS;
    const float* yb = y + b * NPTS;
    for (int e = tid; e < NPTS; e += BLOCK) {
        float xv = xb[e], yv = yb[e];
        mn = fminf(mn, fminf(xv, yv));
        mx = fmaxf(mx, fmaxf(xv, yv));
    }
    smn[tid] = mn; smx[tid] = mx;
    __syncthreads();
    for (int s = BLOCK / 2; s > 0; s >>= 1) {
        if (tid < s) {
            smn[tid] = fminf(smn[tid], smn[tid + s]);
            smx[tid] = fmaxf(smx[tid], smx[tid + s]);
        }
        __syncthreads();
    }
    if (tid == 0) {
        float diam = smx[0] - smn[0];
        float eps0 = diam * diam;
        float* sc  = eps_sched + b * EPS_STRIDE;
        sc[0] = eps0;                         // init softmin at eps0
        float pw = 1.0f;                      // (scaling^2)^max(k-1,0), scaling^2=0.25
        for (int k = 0; k < 16; ++k) {
            sc[1 + k] = fmaxf(BLUR2, eps0 * pw);
            if (k >= 1) pw *= 0.25f;
        }
        sc[17] = BLUR2;                       // final extrapolation step
    }
    for (int e = tid; e < NPTS; e += BLOCK) {
        f0[b * NPTS + e] = 0.0f;
        g0[b * NPTS + e] = 0.0f;
    }
}

// ---------------------------------------------------------------------------
// Kernel 2: one Sinkhorn softmin pair, log2 domain.
//   side 0:  ft_i = -eps*ln2*log2sumexp2_j( h2_j - (log2e/2eps)*(x_i-y_j)^2 )
//   side 1:  symmetric with rows/cols swapped and h from f.
//   avg=1: out = 0.5*(in + t)   (annealing scan step)
//   avg=0: out = t              (init / final extrapolation)
// ---------------------------------------------------------------------------
__global__ __launch_bounds__(BLOCK)
void sink_softmin(const float* __restrict__ x, const float* __restrict__ y,
                  const float* __restrict__ f_in, const float* __restrict__ g_in,
                  float* __restrict__ f_out, float* __restrict__ g_out,
                  const float* __restrict__ eps_sched, int step, int avg)
{
    const int b    = blockIdx.z;
    const int side = blockIdx.y;
    const int tid  = threadIdx.x;

    __shared__ __align__(16) float shS[NPTS];  // column points
    __shared__ __align__(16) float shH[NPTS];  // h2_j = (NLOG + pot_j/eps)*log2e

    const float eps     = eps_sched[b * EPS_STRIDE + step];
    const float inv_eps = 1.0f / eps;

    const float* rowsP = (side == 0) ? x    : y;
    const float* colsP = (side == 0) ? y    : x;
    const float* hsrc  = (side == 0) ? g_in : f_in;
    const float* inP   = (side == 0) ? f_in : g_in;
    float*       outP  = (side == 0) ? f_out : g_out;

    // ---- Stage column points into LDS via async DMA (ASYNCcnt path). ----
    const float* src = colsP + b * NPTS;
    #pragma unroll
    for (int u = 0; u < 3; ++u) {
        int p = tid + u * BLOCK;                       // float-pair index
        unsigned lds = (unsigned)(unsigned long long)(&shS[2 * p]);
        const float* ga = src + 2 * p;
        asm volatile("global_load_async_to_lds_b64 %0, %1, off"
                     :: "v"(lds), "v"(ga) : "memory");
    }
    // ---- h2_j needs the per-element scale anyway -> through VGPRs. ----
    const float hscale = inv_eps * LOG2E;
    const float* hs = hsrc + b * NPTS;
    #pragma unroll
    for (int u = 0; u < 6; ++u) {
        int e = tid + u * BLOCK;
        shH[e] = __builtin_fmaf(hs[e], hscale, NLOG2);
    }
    asm volatile("s_wait_asynccnt 0x0" ::: "memory");   // my async writes landed
    __syncthreads();                                    // everyone's landed

    // ---- One row per thread; two passes; 4 independent accumulators. ----
    const int r  = blockIdx.x * BLOCK + tid;
    const float xv  = rowsP[b * NPTS + r];
    const float nhi = -0.5f * inv_eps * LOG2E;          // log2-domain quadratic coeff

    auto term = [&](int j) -> float {
        float d = xv - shS[j];
        return __builtin_fmaf(d * nhi, d, shH[j]);
    };

    float m0 = -3.0e38f, m1 = -3.0e38f, m2 = -3.0e38f, m3 = -3.0e38f;
    #pragma unroll 2
    for (int j = 0; j < NPTS; j += 4) {
        m0 = fmaxf(m0, term(j + 0));
        m1 = fmaxf(m1, term(j + 1));
        m2 = fmaxf(m2, term(j + 2));
        m3 = fmaxf(m3, term(j + 3));
    }
    const float m = fmaxf(fmaxf(m0, m1), fmaxf(m2, m3));

    float s0 = 0.0f, s1 = 0.0f, s2 = 0.0f, s3 = 0.0f;
    #pragma unroll 2
    for (int j = 0; j < NPTS; j += 4) {
        s0 += __builtin_amdgcn_exp2f(term(j + 0) - m);  // raw v_exp_f32
        s1 += __builtin_amdgcn_exp2f(term(j + 1) - m);
        s2 += __builtin_amdgcn_exp2f(term(j + 2) - m);
        s3 += __builtin_amdgcn_exp2f(term(j + 3) - m);
    }
    const float ssum = (s0 + s1) + (s2 + s3);

    float t = -eps * LN2 * (m + __builtin_amdgcn_logf(ssum));  // v_log_f32 = log2
    if (avg) t = 0.5f * (inP[b * NPTS + r] + t);
    outP[b * NPTS + r] = t;
}

// ---------------------------------------------------------------------------
// Kernel 3a: deterministic strided partial sums -> 512 partials in ws.
// ---------------------------------------------------------------------------
__global__ __launch_bounds__(512)
void sink_partial(const float* __restrict__ f_fin, const float* __restrict__ g_fin,
                  float* __restrict__ part)
{
    const int tid = threadIdx.x;
    float s = 0.0f;
    for (int i = tid; i < BN; i += 512)
        s += f_fin[i] + g_fin[i];
    part[tid] = s;
}

// ---------------------------------------------------------------------------
// Kernel 3b: single wave32 (EXEC all ones, as WMMA requires). Collapse the
// 512 partials to 64 slots (8 adds each, fixed order), then sum all 64 with
// V_WMMA_F32_16X16X4_F32 against a ones B-matrix: D = A(16x4) * ones(4x16),
// so every D column holds the 16 row-sums of A. Lane 0 accumulates rows 0-7,
// lane 16 rows 8-15; shuffle-combine and scale by 1/N.
// ---------------------------------------------------------------------------
__global__ __launch_bounds__(32)
void sink_finalize(const float* __restrict__ part, float* __restrict__ out)
{
    const int tid = threadIdx.x;

    // A-matrix layout (16x4 f32): lanes 0-15 hold K=0,1 in VGPR0/1 for M=lane;
    // lanes 16-31 hold K=2,3 for M=lane-16 => slot id = tid*2 + v, 64 slots.
    float a0 = 0.0f, a1 = 0.0f;
    const int b0 = (tid * 2 + 0) * 8;
    const int b1 = (tid * 2 + 1) * 8;
    #pragma unroll
    for (int k = 0; k < 8; ++k) {
        a0 += part[b0 + k];
        a1 += part[b1 + k];
    }
    v2f a;    a.x = a0;   a.y = a1;
    v2f ones; ones.x = 1.0f; ones.y = 1.0f;
    v8f c = {};
    v8f d = __builtin_amdgcn_wmma_f32_16x16x4_f32(
        /*neg_a=*/false, a, /*neg_b=*/false, ones,
        /*c_mod=*/(short)0, c, /*reuse_a=*/false, /*reuse_b=*/false);

    // column N=lane of D: component r = D[M=r(+8 for lanes>=16)][N]
    float loc = ((d[0] + d[1]) + (d[2] + d[3])) + ((d[4] + d[5]) + (d[6] + d[7]));
    float hi  = __shfl(loc, 16, 32);   // rows 8-15 total lives in lane 16
    if (tid == 0)
        out[0] = (loc + hi) * (1.0f / (float)NPTS);
}

// ---------------------------------------------------------------------------
extern "C" void kernel_launch(void* const* d_in, const int* in_sizes, int n_in,
                              void* d_out, int out_size, void* d_ws, size_t ws_size,
                              hipStream_t stream)
{
    (void)in_sizes; (void)n_in; (void)out_size; (void)ws_size;
    const float* x = (const float*)d_in[0];   // (16, 1536) f32
    const float* y = (const float*)d_in[1];   // (16, 1536) f32

    float* ws   = (float*)d_ws;
    float* fA   = ws;                 // ping/pong potentials
    float* fB   = ws + 1 * BN;
    float* gA   = ws + 2 * BN;
    float* gB   = ws + 3 * BN;
    float* eps  = ws + 4 * BN;        // BATCH * EPS_STRIDE schedule
    float* part = eps + BATCH * EPS_STRIDE;   // 512 partials

    sink_setup<<<BATCH, BLOCK, 0, stream>>>(x, y, fA, gA, eps);

    dim3 grid(RB, 2, BATCH);
    float *fi = fA, *gi = gA, *fo = fB, *go = gB;

    // step 0: init potentials at eps0 (assign; g=f=0 so h = -log N)
    sink_softmin<<<grid, BLOCK, 0, stream>>>(x, y, fi, gi, fo, go, eps, 0, 0);
    { float* t = fi; fi = fo; fo = t; t = gi; gi = go; go = t; }

    // 16 annealing scan steps (parallel update, averaged)
    for (int k = 0; k < 16; ++k) {
        sink_softmin<<<grid, BLOCK, 0, stream>>>(x, y, fi, gi, fo, go, eps, 1 + k, 1);
        { float* t = fi; fi = fo; fo = t; t = gi; gi = go; go = t; }
    }

    // final extrapolation at eps = blur^2 (assign)
    sink_softmin<<<grid, BLOCK, 0, stream>>>(x, y, fi, gi, fo, go, eps, 17, 0);

    sink_partial<<<1, 512, 0, stream>>>(fo, go, part);
    sink_finalize<<<1, 32, 0, stream>>>(part, (float*)d_out);
}